// Foil_8469675508236
// MI455X (gfx1250) — compile-verified
//
#include <hip/hip_runtime.h>

#define THREADS 256
#define TILES_PER_BLOCK 8

#define AS1 __attribute__((address_space(1)))
#define AS3 __attribute__((address_space(3)))

typedef __attribute__((ext_vector_type(4))) int v4i;

// ---- gfx1250 tanh (V_TANH_F32) -------------------------------------------
__device__ __forceinline__ float fast_tanh(float x) {
#if __has_builtin(__builtin_amdgcn_tanhf)
    return __builtin_amdgcn_tanhf(x);
#elif __has_builtin(__builtin_amdgcn_tanh_f32)
    return __builtin_amdgcn_tanh_f32(x);
#else
    return tanhf(x);
#endif
}

// ---- s_wait_asynccnt ------------------------------------------------------
template <int N>
__device__ __forceinline__ void wait_async() {
#if __has_builtin(__builtin_amdgcn_s_wait_asynccnt)
    __builtin_amdgcn_s_wait_asynccnt(N);
#else
    asm volatile("s_wait_asynccnt %0" ::"i"(N) : "memory");
#endif
}

// ---- async global -> LDS stage (GLOBAL_LOAD_ASYNC_TO_LDS_B128) ------------
__device__ __forceinline__ void stage(const float4* __restrict__ src,
                                      float4* ldsDst) {
#if __has_builtin(__builtin_amdgcn_global_load_async_to_lds_b128)
    __builtin_amdgcn_global_load_async_to_lds_b128(
        (AS1 v4i*)src, (AS3 v4i*)ldsDst, 0, 0);
#else
    unsigned lds_addr = (unsigned)(unsigned long long)(AS3 char*)ldsDst;
    asm volatile("global_load_async_to_lds_b128 %0, %1, off"
                 ::"v"(lds_addr), "v"(src)
                 : "memory");
#endif
}

// ---- per-element math: faithful to the reference's degenerate clip --------
// begin == end == 0 (hi == 0), so theta = (1-pos)*T + pos*T with pos = tanh(x)*119
__device__ __forceinline__ float foil(float x, float T, float V) {
    float pos   = fast_tanh(x) * 119.0f;
    float omp   = 1.0f - pos;
    float theta = omp * T + pos * T;
    float velo  = omp * V + pos * V;
    float ds    = velo * 0.01f;
    float dx    = ds * __cosf(theta);
    float dy    = ds * __sinf(theta);
    return x * (1.0f + dy) + dx;
}

__global__ __launch_bounds__(THREADS) void Foil_kernel(
    const float4* __restrict__ in, const float* __restrict__ params,
    float4* __restrict__ out, int n4) {
    __shared__ float4 buf[2][THREADS];

    const int  tid  = threadIdx.x;
    const long base = (long)blockIdx.x * (THREADS * TILES_PER_BLOCK) + tid;

    // params layout: [s][g][p] with strides (480,120,1); only p==0 is ever read
    const float T0 = params[0],   T1 = params[120];
    const float T2 = params[240], T3 = params[360];
    const float V0 = params[480], V1 = params[600];
    const float V2 = params[720], V3 = params[840];

    // prologue: stage tile 0
    if (base < n4) stage(in + base, &buf[0][tid]);

#pragma unroll
    for (int j = 0; j < TILES_PER_BLOCK; ++j) {
        const long idx = base + (long)j * THREADS;

        if (j + 1 < TILES_PER_BLOCK) {
            const long nidx = idx + THREADS;
            if (nidx < n4) stage(in + nidx, &buf[(j + 1) & 1][tid]);
            wait_async<1>();   // tile j landed (async loads complete in order)
        } else {
            wait_async<0>();   // last tile: drain
        }

        float4 x = buf[j & 1][tid];   // ds_load_b128
        float4 r;
        r.x = foil(x.x, T0, V0);      // column%4 == 0 -> group 0
        r.y = foil(x.y, T1, V1);
        r.z = foil(x.z, T2, V2);
        r.w = foil(x.w, T3, V3);

        if (idx < n4) out[idx] = r;   // global_store_b128
    }
}

extern "C" void kernel_launch(void* const* d_in, const int* in_sizes, int n_in,
                              void* d_out, int out_size, void* d_ws, size_t ws_size,
                              hipStream_t stream) {
    const float4* in     = (const float4*)d_in[0];
    const float*  params = (const float*)d_in[1];
    float4*       out    = (float4*)d_out;

    const int n  = in_sizes[0];   // 4096*4096, multiple of 4
    const int n4 = n >> 2;        // float4 count (4,194,304)

    const int elemsPerBlock = THREADS * TILES_PER_BLOCK;          // 2048
    const int blocks = (n4 + elemsPerBlock - 1) / elemsPerBlock;  // 2048

    Foil_kernel<<<blocks, THREADS, 0, stream>>>(in, params, out, n4);
}